// CapsuleNet_49967649521769
// MI455X (gfx1250) — compile-verified
//
#include <hip/hip_runtime.h>
#include <math.h>
#include <stdint.h>

// Sizes (from reference): B=4, GL=4, GF=128, N=1024, CS=32, CN=16, NA=16, S=512, HS=512
// R = GL*CS = 128 primary rows, K1 = GL*GF = 512, Q = CN*CS = 512, KK = GL*CS = 128

typedef float v2f __attribute__((ext_vector_type(2)));
typedef float v8f __attribute__((ext_vector_type(8)));

__device__ __forceinline__ v8f wmma4(v2f a, v2f b, v8f c) {
  // D = A(16x4 f32) * B(4x16 f32) + C(16x16 f32)  -> v_wmma_f32_16x16x4_f32
  return __builtin_amdgcn_wmma_f32_16x16x4_f32(false, a, false, b, (short)0, c,
                                               false, false);
}

// Async copy of 16 contiguous bytes global->LDS (ASYNCcnt-tracked, no VGPR data).
__device__ __forceinline__ void async_copy_b128(uint32_t lds_byte_off,
                                                const void* gptr) {
  asm volatile("global_load_async_to_lds_b128 %0, %1, off"
               :: "v"(lds_byte_off), "v"((uint64_t)gptr)
               : "memory");
}
__device__ __forceinline__ void wait_asynccnt0() {
  asm volatile("s_wait_asynccnt 0x0" ::: "memory");
}

// ---------------- K1: primary capsules: U[b][128][1024] = Wp[128x512] * X[512x1024] + bp
__global__ __launch_bounds__(32)
void k1_primary_gemm(const float* __restrict__ Wp, const float* __restrict__ X,
                     const float* __restrict__ bp, float* __restrict__ U) {
  const int lane = threadIdx.x;
  const int nt = blockIdx.x;   // 0..63 column tiles
  const int rt = blockIdx.y;   // 0..7 row tiles
  const int b  = blockIdx.z;
  const int l15 = lane & 15;
  const int khalf = (lane >> 4) * 2;           // 0 or 2 (A/B K placement per ISA layout)
  const int arow = rt * 16 + l15;
  const int bcol = nt * 16 + l15;
  const float* Xb = X + b * (512 * 1024);
  v8f acc = {};
  for (int k = 0; k < 512; k += 4) {
    if (k + 16 < 512)  // warm L2/WGP$ ahead of the streamed B operand
      __builtin_prefetch(Xb + (k + 16 + khalf) * 1024 + bcol, 0, 1);
    v2f a = *(const v2f*)(Wp + arow * 512 + k + khalf);      // K, K+1 contiguous
    v2f bb;
    bb.x = Xb[(k + khalf + 0) * 1024 + bcol];
    bb.y = Xb[(k + khalf + 1) * 1024 + bcol];
    acc = wmma4(a, bb, acc);
  }
  float* Ub = U + b * (128 * 1024);
  const int rbase = rt * 16 + ((lane >> 4) ? 8 : 0);
#pragma unroll
  for (int r = 0; r < 8; ++r) {
    const int row = rbase + r;
    Ub[row * 1024 + bcol] = acc[r] + bp[row];
  }
}

// ---------------- K2a: mag_sq per (b,g) over CS*N = 32768 contiguous elements
__global__ __launch_bounds__(256)
void k2a_mag(const float* __restrict__ U, float* __restrict__ magsq) {
  __shared__ float red[256];
  const int bg = blockIdx.x;                 // b*4+g, 16 blocks
  const float* base = U + bg * 32768;
  float s = 0.f;
  for (int i = threadIdx.x; i < 32768; i += 256) { float v = base[i]; s += v * v; }
  red[threadIdx.x] = s;
  __syncthreads();
  for (int o = 128; o > 0; o >>= 1) {
    if (threadIdx.x < o) red[threadIdx.x] += red[threadIdx.x + o];
    __syncthreads();
  }
  if (threadIdx.x == 0) magsq[bg] = red[0];
}

// ---------------- K2b: squash scale in place
__global__ __launch_bounds__(256)
void k2b_scale(float* __restrict__ U, const float* __restrict__ magsq) {
  const int idx = blockIdx.x * 256 + threadIdx.x;  // < 524288
  const float m = magsq[idx >> 15];
  U[idx] *= sqrtf(m) / (1.0f + m);
}

__global__ void k_zero(float* __restrict__ p, int n) {
  const int i = blockIdx.x * blockDim.x + threadIdx.x;
  if (i < n) p[i] = 0.f;
}

// ---------------- Kc: c[b][i][j] = softmax_j(blog[b][i][j]),  i in CS=32
__global__ __launch_bounds__(128)
void kc_softmax(const float* __restrict__ blog, float* __restrict__ c) {
  const int t = threadIdx.x;          // 128 = B*32
  const int b = t >> 5, i = t & 31;
  const float* row = blog + b * 512 + i * 16;
  float mx = row[0];
  for (int j = 1; j < 16; ++j) mx = fmaxf(mx, row[j]);
  float e[16], sum = 0.f;
  for (int j = 0; j < 16; ++j) { e[j] = __expf(row[j] - mx); sum += e[j]; }
  const float inv = 1.f / sum;
  float* out = c + b * 512 + i * 16;
  for (int j = 0; j < 16; ++j) out[j] = e[j] * inv;
}

// ---------------- Kw: Weff[b][(k*32+i)][(j*32+u)] = c[b][i][j] * Wg[i][j][u][k]
__global__ __launch_bounds__(256)
void kw_build(const float* __restrict__ c, const float* __restrict__ Wg,
              float* __restrict__ Weff) {
  const int idx = blockIdx.x * 256 + threadIdx.x;  // < B*65536
  const int b = idx >> 16;
  const int t = idx & 65535;
  const int kk = t >> 9, q = t & 511;
  const int k = kk >> 5, i = kk & 31;
  const int j = q >> 5, u = q & 31;
  Weff[idx] = c[b * 512 + i * 16 + j] * Wg[((i * 16 + j) * 32 + u) * 4 + k];
}

// ---------------- Ks: s[b][1024][512] = U(p)[1024x128] * Weff[128x512]   (WMMA)
// 8 waves/block share one q-tile: the 128x16 B-panel (8 KB) is staged into LDS
// once via global_load_async_to_lds_b128, then each wave streams a different
// m-tile of A (contiguous v2f) against the LDS-resident panel.
__global__ __launch_bounds__(256)
void ks_gemm(const float* __restrict__ U, const float* __restrict__ Weff,
             float* __restrict__ sbuf) {
  __shared__ float bsh[128 * 16];          // B panel, row-major [k][col]
  const int tid = threadIdx.x;
  const int qt = blockIdx.x;   // 0..31
  const int mtg = blockIdx.y;  // 0..7 (8 m-tiles per block)
  const int b = blockIdx.z;
  const float* Bm = Weff + b * 65536 + qt * 16;   // row k at Bm + k*512

  // cooperative async stage: 2048 floats = 512 x b128 lane-transfers
#pragma unroll
  for (int i = 0; i < 2; ++i) {
    const int f = (i * 256 + tid) * 4;     // float index in panel, 16B chunks
    const int k = f >> 4, cc = f & 15;
    async_copy_b128((uint32_t)(uintptr_t)&bsh[f], Bm + k * 512 + cc);
  }
  wait_asynccnt0();
  __syncthreads();

  const int lane = tid & 31, wid = tid >> 5;
  const int l15 = lane & 15;
  const int khalf = (lane >> 4) * 2;
  const int mt = mtg * 8 + wid;
  const float* A = U + b * 131072;         // [1024,128]
  const int arow = mt * 16 + l15;
  v8f acc = {};
  for (int k = 0; k < 128; k += 4) {
    v2f a = *(const v2f*)(A + arow * 128 + k + khalf);
    v2f bb;
    bb.x = bsh[(k + khalf + 0) * 16 + l15];
    bb.y = bsh[(k + khalf + 1) * 16 + l15];
    acc = wmma4(a, bb, acc);
  }
  float* Sb = sbuf + b * 524288;
  const int bcol = qt * 16 + l15;
  const int rbase = mt * 16 + ((lane >> 4) ? 8 : 0);
#pragma unroll
  for (int r = 0; r < 8; ++r) Sb[(rbase + r) * 512 + bcol] = acc[r];
}

// ---------------- Kv: squash over J (16) per (b,m,u), in place
__global__ __launch_bounds__(256)
void kv_squash(float* __restrict__ sbuf) {
  const int idx = blockIdx.x * 256 + threadIdx.x;  // < B*1024*32
  const int b = idx >> 15;
  const int t = idx & 32767;
  const int m = t >> 5, u = t & 31;
  float* base = sbuf + b * 524288 + m * 512 + u;
  float v[16], mag = 0.f;
#pragma unroll
  for (int j = 0; j < 16; ++j) { v[j] = base[j * 32]; mag += v[j] * v[j]; }
  const float sc = sqrtf(mag) / (1.0f + mag);
#pragma unroll
  for (int j = 0; j < 16; ++j) base[j * 32] = v[j] * sc;
}

// ---------------- Kt: T[b][128][512] = (1/1024) * U^T[128x1024] * v[1024x512]  (WMMA)
__global__ __launch_bounds__(32)
void kt_gemm(const float* __restrict__ U, const float* __restrict__ vbuf,
             float* __restrict__ T) {
  const int lane = threadIdx.x;
  const int qt = blockIdx.x;   // 0..31
  const int pt = blockIdx.y;   // 0..7
  const int b = blockIdx.z;
  const float* A = U + b * 131072;       // [1024,128] accessed transposed
  const float* V = vbuf + b * 524288;    // [1024,512]
  const int l15 = lane & 15;
  const int khalf = (lane >> 4) * 2;
  const int prow = pt * 16 + l15;
  const int qcol = qt * 16 + l15;
  v8f acc = {};
  for (int m = 0; m < 1024; m += 4) {
    v2f a, bb;
    a.x = A[(m + khalf + 0) * 128 + prow];
    a.y = A[(m + khalf + 1) * 128 + prow];
    bb.x = V[(m + khalf + 0) * 512 + qcol];
    bb.y = V[(m + khalf + 1) * 512 + qcol];
    acc = wmma4(a, bb, acc);
  }
  float* Tb = T + b * 65536;
  const int rbase = pt * 16 + ((lane >> 4) ? 8 : 0);
  const float scale = 1.0f / 1024.0f;
#pragma unroll
  for (int r = 0; r < 8; ++r) Tb[(rbase + r) * 512 + qcol] = acc[r] * scale;
}

// ---------------- Kb: blog[b][i][j] += sum_{k,u} Wg[i][j][u][k] * T[b][(k*32+i)][(j*32+u)]
__global__ __launch_bounds__(256)
void kb_update(const float* __restrict__ Wg, const float* __restrict__ T,
               float* __restrict__ blog) {
  const int idx = blockIdx.x * 256 + threadIdx.x;  // < 2048
  const int b = idx >> 9;
  const int t = idx & 511;
  const int i = t >> 4, j = t & 15;
  const float* Tb = T + b * 65536;
  float acc = 0.f;
  for (int k = 0; k < 4; ++k)
    for (int u = 0; u < 32; ++u)
      acc += Wg[((i * 16 + j) * 32 + u) * 4 + k] * Tb[(k * 32 + i) * 512 + j * 32 + u];
  blog[b * 512 + i * 16 + j] += acc;
}

// ---------------- Kg: g[b][q] = mean_m v[b][m][q]
__global__ __launch_bounds__(256)
void kg_mean(const float* __restrict__ vbuf, float* __restrict__ g) {
  __shared__ float red[256];
  const int bq = blockIdx.x;  // 2048 blocks
  const int b = bq >> 9, q = bq & 511;
  const float* base = vbuf + b * 524288 + q;
  float s = 0.f;
  for (int m = threadIdx.x; m < 1024; m += 256) s += base[m * 512];
  red[threadIdx.x] = s;
  __syncthreads();
  for (int o = 128; o > 0; o >>= 1) {
    if (threadIdx.x < o) red[threadIdx.x] += red[threadIdx.x + o];
    __syncthreads();
  }
  if (threadIdx.x == 0) g[bq] = red[0] * (1.0f / 1024.0f);
}

// ---------------- Ka: attention + aspect routing (M collapses to 1 by softmax
// shift-invariance: score[s][j] = softmax_j(gdot[j] + hdot[s]) == softmax_j(gdot[j]))
__global__ __launch_bounds__(256)
void ka_aspect(const float* __restrict__ g, const float* __restrict__ Wa,
               const float* __restrict__ Ws, float* __restrict__ va) {
  __shared__ float gs[512];
  __shared__ float score[16];
  __shared__ float cond[512];
  __shared__ float uh[16 * 16 * 32];   // u_hat_a[i][j][u]
  __shared__ float blog[256];          // b_a[i][j]
  __shared__ float cc[256];            // c[i][j]
  __shared__ float sv[512];            // s / v [j][u]
  const int b = blockIdx.x;
  const int tid = threadIdx.x;
  for (int t = tid; t < 512; t += 256) gs[t] = g[b * 512 + t];
  blog[tid] = 0.f;
  __syncthreads();
  if (tid == 0) {
    float gd[16], mx = -1e30f;
    for (int j = 0; j < 16; ++j) {
      float s = 0.f;
      for (int u = 0; u < 32; ++u) s += gs[j * 32 + u] * Wa[u];
      gd[j] = s; mx = fmaxf(mx, s);
    }
    float sum = 0.f;
    for (int j = 0; j < 16; ++j) { gd[j] = __expf(gd[j] - mx); sum += gd[j]; }
    const float inv = 1.f / sum;
    for (int j = 0; j < 16; ++j) score[j] = gd[j] * inv;
  }
  __syncthreads();
  for (int t = tid; t < 512; t += 256) cond[t] = gs[t] * score[t >> 5];
  __syncthreads();
  // u_hat_a[i][j][u] = sum_k Ws[i][j][u][k] * cond[i][k]
  for (int t = tid; t < 8192; t += 256) {
    const int i = t >> 9;
    const float* wrow = Ws + t * 32;
    const float* crow = cond + i * 32;
    float s = 0.f;
    for (int k = 0; k < 32; ++k) s += wrow[k] * crow[k];
    uh[t] = s;
  }
  __syncthreads();
  for (int it = 0; it < 3; ++it) {
    if (tid < 16) {                     // softmax rows of blog
      const int i = tid;
      float mx = blog[i * 16];
      for (int j = 1; j < 16; ++j) mx = fmaxf(mx, blog[i * 16 + j]);
      float sum = 0.f;
      for (int j = 0; j < 16; ++j) { float e = __expf(blog[i * 16 + j] - mx); cc[i * 16 + j] = e; sum += e; }
      const float inv = 1.f / sum;
      for (int j = 0; j < 16; ++j) cc[i * 16 + j] *= inv;
    }
    __syncthreads();
    for (int t = tid; t < 512; t += 256) {    // s[j][u] = sum_i c[i][j]*uh[i][j][u]
      const int j = t >> 5;
      float s = 0.f;
      for (int i = 0; i < 16; ++i) s += cc[i * 16 + j] * uh[i * 512 + t];
      sv[t] = s;
    }
    __syncthreads();
    if (tid < 32) {                     // squash over J per u
      const int u = tid;
      float mag = 0.f;
      for (int j = 0; j < 16; ++j) { float x = sv[j * 32 + u]; mag += x * x; }
      const float sc = sqrtf(mag) / (1.0f + mag);
      for (int j = 0; j < 16; ++j) sv[j * 32 + u] *= sc;
    }
    __syncthreads();
    if (it < 2) {
      if (tid < 256) {                  // blog[i][j] += sum_u uh[i][j][u]*v[j][u]
        const int i = tid >> 4, j = tid & 15;
        float s = 0.f;
        for (int u = 0; u < 32; ++u) s += uh[(i * 16 + j) * 32 + u] * sv[j * 32 + u];
        blog[tid] += s;
      }
      __syncthreads();
    }
  }
  for (int t = tid; t < 512; t += 256) va[b * 512 + t] = sv[t];
}

// ---------------- Ko: broadcast v_a to all S=512 positions: out[b][s][j][u]
__global__ __launch_bounds__(256)
void ko_broadcast(const float* __restrict__ va, float* __restrict__ out) {
  const int idx = blockIdx.x * 256 + threadIdx.x;  // < 4*512*512
  const int b = idx >> 18;
  const int q = idx & 511;
  out[idx] = va[b * 512 + q];
}

extern "C" void kernel_launch(void* const* d_in, const int* in_sizes, int n_in,
                              void* d_out, int out_size, void* d_ws, size_t ws_size,
                              hipStream_t stream) {
  (void)in_sizes; (void)n_in; (void)out_size; (void)ws_size;
  const float* graph_embed = (const float*)d_in[0];
  // d_in[1] = hidden: mathematically unused (softmax shift-invariance)
  const float* Wp = (const float*)d_in[2];
  const float* bp = (const float*)d_in[3];
  const float* Wg = (const float*)d_in[4];
  const float* Wa = (const float*)d_in[5];
  const float* Ws = (const float*)d_in[6];
  float* out = (float*)d_out;

  float* ws    = (float*)d_ws;
  float* U     = ws;                 // 524288 : primary caps / p rows [b][m*128+kk]
  float* magsq = U + 524288;         // 16
  float* blog  = magsq + 16;         // 2048  : graph routing logits
  float* cbuf  = blog + 2048;        // 2048
  float* Weff  = cbuf + 2048;        // 262144
  float* sbuf  = Weff + 262144;      // 2097152 : s / v [b][m][q]
  float* T     = sbuf + 2097152;     // 262144
  float* g     = T + 262144;         // 2048
  float* va    = g + 2048;           // 2048

  k1_primary_gemm<<<dim3(64, 8, 4), 32, 0, stream>>>(Wp, graph_embed, bp, U);
  k2a_mag<<<16, 256, 0, stream>>>(U, magsq);
  k2b_scale<<<2048, 256, 0, stream>>>(U, magsq);
  k_zero<<<8, 256, 0, stream>>>(blog, 2048);
  for (int it = 0; it < 3; ++it) {
    kc_softmax<<<1, 128, 0, stream>>>(blog, cbuf);
    kw_build<<<1024, 256, 0, stream>>>(cbuf, Wg, Weff);
    ks_gemm<<<dim3(32, 8, 4), 256, 0, stream>>>(U, Weff, sbuf);
    kv_squash<<<512, 256, 0, stream>>>(sbuf);
    if (it < 2) {
      kt_gemm<<<dim3(32, 8, 4), 32, 0, stream>>>(U, sbuf, T);
      kb_update<<<8, 256, 0, stream>>>(Wg, T, blog);
    }
  }
  kg_mean<<<2048, 256, 0, stream>>>(sbuf, g);
  ka_aspect<<<4, 256, 0, stream>>>(g, Wa, Ws, va);
  ko_broadcast<<<4096, 256, 0, stream>>>(va, out);
}